// NeuralDisCoCirc_88794153877945
// MI455X (gfx1250) — compile-verified
//
#include <hip/hip_runtime.h>
#include <hip/hip_bf16.h>

typedef __attribute__((ext_vector_type(2))) float v2f;
typedef __attribute__((ext_vector_type(8))) float v8f;

#define NL   8
#define BSZ  16
#define DSZ  1024

// One diagram layer: h_out[b, j] = maskedReLU( sum_i h_in[b,i] * W[b,i,j] + bias[b,j] )
// Grid: (DSZ/32, BSZ) blocks of 256 threads (8 waves).
// Wave w: column subtile (w&1)*16 within the block's 32-col tile, K-quarter (w>>1).
//
// A-fragment trick: every lane carries the same broadcast h values, so ALL
// rows of A equal h[k..k+3]. Rows of a WMMA are independent, so every D row
// computes the same (correct) dot product; we read only row 0. This avoids
// any lane-select / exec-mask manipulation in the hot loop.
__global__ __launch_bounds__(256) void discocirc_layer_kernel(
    const float* __restrict__ h_in,   // [B, D]
    const float* __restrict__ W,      // [B, D, D] (this layer)
    const float* __restrict__ bias,   // [B, D]   (this layer)
    const int*   __restrict__ mask,   // [B, D]   (this layer)
    float*       __restrict__ h_out)  // [B, D]
{
    __shared__ float lds_h[DSZ];
    __shared__ float lds_acc[32];

    const int b      = blockIdx.y;
    const int jt     = blockIdx.x;        // 32-column tile
    const int tid    = threadIdx.x;
    const int lane   = tid & 31;
    const int wave   = tid >> 5;          // 0..7
    const int laneLo = lane & 15;
    const int hiHalf = lane >> 4;         // 0 or 1

    // Stage this sample's activation row into LDS once.
    for (int i = tid; i < DSZ; i += 256) lds_h[i] = h_in[b * DSZ + i];
    if (tid < 32) lds_acc[tid] = 0.0f;
    __syncthreads();

    const int colSub   = wave & 1;        // which 16-col subtile
    const int kQuarter = wave >> 1;       // 0..3 -> K range of 256
    const int j0 = jt * 32 + colSub * 16;
    const int k0 = kQuarter * 256;

    // Base pointer for this lane's B-fragment loads; all loads inside a
    // 4-WMMA chunk are reached via constant immediate offsets so the
    // compiler emits one s_clause of 8 global_load_b32 per chunk.
    const float* __restrict__ p =
        W + (size_t)b * DSZ * DSZ + (size_t)(k0 + 2 * hiHalf) * DSZ + (j0 + laneLo);
    const float* __restrict__ hp = lds_h + k0 + 2 * hiHalf;

    v8f c0 = {0.f, 0.f, 0.f, 0.f, 0.f, 0.f, 0.f, 0.f};
    v8f c1 = {0.f, 0.f, 0.f, 0.f, 0.f, 0.f, 0.f, 0.f};

    // 64 WMMAs total: 16 outer iterations x 4 WMMAs (k advances 16 per iter).
    #pragma unroll 2
    for (int it = 0; it < 16; ++it) {
        // A fragments: unconditional LDS broadcast loads (same value in all
        // lanes of each half-wave). Lanes 0-15 carry K={0,1}, lanes 16-31
        // carry K={2,3}.
        v2f a0, a1, a2, a3;
        a0.x = hp[0];   a0.y = hp[1];
        a1.x = hp[4];   a1.y = hp[5];
        a2.x = hp[8];   a2.y = hp[9];
        a3.x = hp[12];  a3.y = hp[13];

        // B fragments (4x16 each): VGPR0 = rows {r, r+2}, VGPR1 = rows {r+1, r+3}.
        // Constant float offsets: +1024 = next row, +4096 = next 4-row chunk.
        v2f b0, b1, b2, b3;
        b0.x = p[0];      b0.y = p[1024];
        b1.x = p[4096];   b1.y = p[5120];
        b2.x = p[8192];   b2.y = p[9216];
        b3.x = p[12288];  b3.y = p[13312];

        c0 = __builtin_amdgcn_wmma_f32_16x16x4_f32(false, a0, false, b0, (short)0, c0, false, false);
        c1 = __builtin_amdgcn_wmma_f32_16x16x4_f32(false, a1, false, b1, (short)0, c1, false, false);
        c0 = __builtin_amdgcn_wmma_f32_16x16x4_f32(false, a2, false, b2, (short)0, c0, false, false);
        c1 = __builtin_amdgcn_wmma_f32_16x16x4_f32(false, a3, false, b3, (short)0, c1, false, false);

        p  += 16384;   // 16 rows of 1024 floats
        hp += 16;
    }

    // D[M=0, N=laneLo] is element 0 on lanes 0..15. Reduce K-quarters across waves.
    const float part = c0[0] + c1[0];
    if (lane < 16) atomicAdd(&lds_acc[colSub * 16 + lane], part);
    __syncthreads();

    if (tid < 32) {
        const int j = jt * 32 + tid;
        float v = lds_acc[tid] + bias[b * DSZ + j];
        const int m = mask[b * DSZ + j];
        h_out[b * DSZ + j] = m ? fmaxf(v, 0.0f) : v;
    }
}

extern "C" void kernel_launch(void* const* d_in, const int* in_sizes, int n_in,
                              void* d_out, int out_size, void* d_ws, size_t ws_size,
                              hipStream_t stream) {
    const float* x       = (const float*)d_in[0];   // [B, D]
    const float* weights = (const float*)d_in[1];   // [L, B, D, D]
    const float* biases  = (const float*)d_in[2];   // [L, B, D]
    const int*   masks   = (const int*)  d_in[3];   // [L, B, D]
    float* out = (float*)d_out;

    // Ping-pong activation buffers in workspace (2 * B * D floats = 128 KB).
    float* buf0 = (float*)d_ws;
    float* buf1 = buf0 + BSZ * DSZ;

    dim3 grid(DSZ / 32, BSZ);
    const float* cur = x;
    for (int l = 0; l < NL; ++l) {
        float* o = (l == NL - 1) ? out : ((l & 1) ? buf1 : buf0);
        discocirc_layer_kernel<<<grid, 256, 0, stream>>>(
            cur,
            weights + (size_t)l * BSZ * DSZ * DSZ,
            biases  + (size_t)l * BSZ * DSZ,
            masks   + (size_t)l * BSZ * DSZ,
            o);
        cur = o;
    }
}